// MambaSSM_15401752723577
// MI455X (gfx1250) — compile-verified
//
#include <hip/hip_runtime.h>
#include <math.h>

#define D_MODEL 1024
#define D_STATE 16
#define D_INNER 2048
#define IN_DIM  256
#define OUT_DIM 256
#define BATCH   2
#define TLEN    1024
#define NROW    (BATCH*TLEN)          // 2048 rows
#define W_X_N   (2*D_STATE + D_INNER) // 2080

typedef __attribute__((ext_vector_type(2))) float v2f;
typedef __attribute__((ext_vector_type(8))) float v8f;

__device__ __forceinline__ float siluf(float x)     { return x / (1.f + __expf(-x)); }
__device__ __forceinline__ float softplusf(float x) { return fmaxf(x, 0.f) + log1pf(__expf(-fabsf(x))); }

// ---------------------------------------------------------------------------
// fp32 WMMA GEMM: C[M,N] = A[M,K] @ B[K,N] (+bias) (+softplus)
// One wave32 computes a 16 x (16*TN) tile: TN V_WMMA_F32_16X16X4_F32
// accumulators share one A fragment per K-step of 4.
// Layouts per CDNA5 ISA 7.12.2:
//   A 16x4 : lane half selects K-pair {0,1}/{2,3}; VGPR j holds K=2*half+j; m=lane&15
//   B 4x16 : mirrored: lane half selects K-pair; n=lane&15
//   C/D    : VGPR i -> M = half*8 + i, N = lane&15
// All addressing is pointer-bump (no per-iteration 64-bit mul); prefetch is
// unconditional (speculative prefetch drops on translation failure).
// ---------------------------------------------------------------------------
template <int TN>
__global__ __launch_bounds__(256)
void gemm_wmma_f32(const float* __restrict__ A, int lda,
                   const float* __restrict__ Bm, int ldb,
                   float* __restrict__ C, int ldc,
                   int M, int N, int K,
                   const float* __restrict__ bias, int act)
{
    const int lane = threadIdx.x & 31;
    const int wave = threadIdx.x >> 5;
    const int tiles_n = N / (16 * TN);
    const int total   = (M >> 4) * tiles_n;
    const int tile = blockIdx.x * 8 + wave;     // wave-uniform guard
    if (tile >= total) return;
    const int tm   = tile / tiles_n;
    const int tn   = tile % tiles_n;
    const int half = lane >> 4;
    const int l16  = lane & 15;

    const size_t ldbs = (size_t)ldb;
    const float* __restrict__ ap = A  + (size_t)(tm * 16 + l16) * lda + 2 * half;
    const float* __restrict__ bp = Bm + (size_t)(2 * half) * ldbs + tn * (16 * TN) + l16;

    v8f acc[TN];
#pragma unroll
    for (int j = 0; j < TN; ++j) acc[j] = (v8f){};

    for (int k = 0; k < K; k += 4) {
        __builtin_prefetch(ap + 16, 0, 1);            // global_prefetch_b8 (A stream)
        __builtin_prefetch(bp + 8 * ldbs, 0, 1);      // global_prefetch_b8 (B stream)

        v2f a;
        a.x = ap[0];
        a.y = ap[1];

#pragma unroll
        for (int j = 0; j < TN; ++j) {
            v2f b;
            b.x = bp[j * 16];
            b.y = bp[ldbs + j * 16];
            // 8 args: (neg_a, A, neg_b, B, c_mod, C, reuse_a, reuse_b)
            acc[j] = __builtin_amdgcn_wmma_f32_16x16x4_f32(false, a, false, b,
                                                           (short)0, acc[j],
                                                           false, false);
        }
        ap += 4;
        bp += 4 * ldbs;
    }

    const int nb = tn * (16 * TN) + l16;
#pragma unroll
    for (int j = 0; j < TN; ++j) {
        const int n = nb + j * 16;
        const float bb = bias ? bias[n] : 0.f;
#pragma unroll
        for (int i = 0; i < 8; ++i) {
            const int m = tm * 16 + half * 8 + i;
            float v = acc[j][i] + bb;
            if (act == 1) v = softplusf(v);
            C[(size_t)m * ldc + n] = v;
        }
    }
}

// ---------------------------------------------------------------------------
// LayerNorm over D_MODEL=1024, one block (8 wave32s) per row, in place.
// ---------------------------------------------------------------------------
__global__ __launch_bounds__(256)
void layernorm_kernel(float* __restrict__ h,
                      const float* __restrict__ gamma,
                      const float* __restrict__ beta)
{
    __shared__ float red[16];
    float* p = h + (size_t)blockIdx.x * D_MODEL;
    float s = 0.f, s2 = 0.f;
    for (int i = threadIdx.x; i < D_MODEL; i += 256) {
        float v = p[i];
        s += v; s2 += v * v;
    }
    for (int o = 16; o > 0; o >>= 1) {
        s  += __shfl_down(s, o, 32);
        s2 += __shfl_down(s2, o, 32);
    }
    const int wv = threadIdx.x >> 5, ln = threadIdx.x & 31;
    if (ln == 0) { red[wv] = s; red[8 + wv] = s2; }
    __syncthreads();
    if (threadIdx.x == 0) {
        float ts = 0.f, ts2 = 0.f;
        for (int w = 0; w < 8; ++w) { ts += red[w]; ts2 += red[8 + w]; }
        red[0] = ts; red[1] = ts2;
    }
    __syncthreads();
    const float mean = red[0] * (1.f / D_MODEL);
    const float var  = red[1] * (1.f / D_MODEL) - mean * mean;
    const float inv  = rsqrtf(var + 1e-5f);
    for (int i = threadIdx.x; i < D_MODEL; i += 256)
        p[i] = (p[i] - mean) * inv * gamma[i] + beta[i];
}

// ---------------------------------------------------------------------------
// Depthwise causal conv (D_CONV=4) over T + SiLU. x_branch = xz[:, :, 0:2048].
// ---------------------------------------------------------------------------
__global__ __launch_bounds__(256)
void conv_silu_kernel(const float* __restrict__ xz,
                      const float* __restrict__ cw,
                      const float* __restrict__ cb,
                      float* __restrict__ u)
{
    const int idx = blockIdx.x * 256 + threadIdx.x;   // < NROW*D_INNER
    const int d   = idx & (D_INNER - 1);
    const int row = idx >> 11;                        // b*T + t
    const int t   = row & (TLEN - 1);
    float acc = cb[d];
    const float* w = cw + d * 4;
#pragma unroll
    for (int k = 0; k < 4; ++k) {
        const int tt = t - 3 + k;
        if (tt >= 0)
            acc += w[k] * xz[(size_t)(row - 3 + k) * (2 * D_INNER) + d];
    }
    u[idx] = siluf(acc);
}

// ---------------------------------------------------------------------------
// Selective scan. Thread = (b, d, n); 16-lane groups reduce over n via
// __shfl_xor (wave32). 2048 waves in flight hide the per-step exp latency.
// ---------------------------------------------------------------------------
__global__ __launch_bounds__(256)
void scan_kernel(const float* __restrict__ dt, const float* __restrict__ u,
                 const float* __restrict__ proj, const float* __restrict__ A_log,
                 const float* __restrict__ h0, float* __restrict__ y,
                 float* __restrict__ hT)
{
    const int idx = blockIdx.x * 256 + threadIdx.x;   // 0 .. B*D_INNER*16-1
    const int n = idx & 15;
    const int d = (idx >> 4) & (D_INNER - 1);
    const int b = idx >> 15;

    const float An = -__expf(A_log[(d << 4) + n]);
    float h = h0[idx];                                 // (B, D_INNER, 16) flat == idx

    const float* dtp = dt   + (size_t)b * TLEN * D_INNER + d;
    const float* up  = u    + (size_t)b * TLEN * D_INNER + d;
    const float* pp  = proj + (size_t)b * TLEN * W_X_N;
    float*       yp  = y    + (size_t)b * TLEN * D_INNER + d;

    for (int t = 0; t < TLEN; ++t) {
        const float dtv = dtp[(size_t)t * D_INNER];
        const float uv  = up [(size_t)t * D_INNER];
        const float Bv  = pp [(size_t)t * W_X_N + n];
        const float Cv  = pp [(size_t)t * W_X_N + D_STATE + n];
        const float dA  = __expf(dtv * An);
        h = dA * h + dtv * Bv * uv;
        float c = h * Cv;
        c += __shfl_xor(c, 1, 32);
        c += __shfl_xor(c, 2, 32);
        c += __shfl_xor(c, 4, 32);
        c += __shfl_xor(c, 8, 32);
        if (n == 0) yp[(size_t)t * D_INNER] = c;
    }
    hT[idx] = h;
}

// ---------------------------------------------------------------------------
// yw = (y + u*D_skip) * silu(z),  z = xz[:, :, 2048:4096]
// ---------------------------------------------------------------------------
__global__ __launch_bounds__(256)
void gate_kernel(const float* __restrict__ y, const float* __restrict__ u,
                 const float* __restrict__ xz, const float* __restrict__ Dskip,
                 float* __restrict__ yw)
{
    const int idx = blockIdx.x * 256 + threadIdx.x;
    const int d   = idx & (D_INNER - 1);
    const int row = idx >> 11;
    const float z = xz[(size_t)row * (2 * D_INNER) + D_INNER + d];
    yw[idx] = (y[idx] + u[idx] * Dskip[d]) * siluf(z);
}

// ---------------------------------------------------------------------------
extern "C" void kernel_launch(void* const* d_in, const int* in_sizes, int n_in,
                              void* d_out, int out_size, void* d_ws, size_t ws_size,
                              hipStream_t stream)
{
    (void)in_sizes; (void)n_in; (void)out_size; (void)ws_size;

    const float* x          = (const float*)d_in[0];
    const float* h0         = (const float*)d_in[1];
    const float* W_in_wrap  = (const float*)d_in[2];
    const float* b_in_wrap  = (const float*)d_in[3];
    const float* ln_gamma   = (const float*)d_in[4];
    const float* ln_beta    = (const float*)d_in[5];
    const float* W_in       = (const float*)d_in[6];
    const float* conv_w     = (const float*)d_in[7];
    const float* conv_b     = (const float*)d_in[8];
    const float* W_x        = (const float*)d_in[9];
    const float* W_dt       = (const float*)d_in[10];
    const float* b_dt       = (const float*)d_in[11];
    const float* A_log      = (const float*)d_in[12];
    const float* D_skip     = (const float*)d_in[13];
    const float* W_out      = (const float*)d_in[14];
    const float* W_out_wrap = (const float*)d_in[15];
    const float* b_out_wrap = (const float*)d_in[16];

    float* out = (float*)d_out;                    // (B,T,OUT_DIM)
    float* hT  = out + (size_t)NROW * OUT_DIM;     // tuple part 2: (B,D_INNER,16)

    float* ws = (float*)d_ws;
    size_t o = 0;
    float* h_pre = ws + o; o += (size_t)NROW * D_MODEL;       // 2M f
    float* xz    = ws + o; o += (size_t)NROW * 2 * D_INNER;   // 8M f
    float* u     = ws + o; o += (size_t)NROW * D_INNER;       // 4M f
    float* proj  = ws + o; o += (size_t)NROW * W_X_N;         // 4.06M f
    float* dtb   = ws + o; o += (size_t)NROW * D_INNER;       // 4M f
    float* yv    = ws + o; o += (size_t)NROW * D_INNER;       // 4M f  (~104 MB total)
    float* yw = dtb;    // dt dead after scan
    float* t1 = h_pre;  // h dead after xz GEMM

    auto gemm = [&](const float* A, int lda, const float* B, int ldb,
                    float* C, int ldc, int M, int N, int K,
                    const float* bias, int act) {
        if ((N & 63) == 0) {
            const int waves  = (M / 16) * (N / 64);
            const int blocks = (waves + 7) / 8;
            hipLaunchKernelGGL((gemm_wmma_f32<4>), dim3(blocks), dim3(256), 0, stream,
                               A, lda, B, ldb, C, ldc, M, N, K, bias, act);
        } else {
            const int waves  = (M / 16) * (N / 32);
            const int blocks = (waves + 7) / 8;
            hipLaunchKernelGGL((gemm_wmma_f32<2>), dim3(blocks), dim3(256), 0, stream,
                               A, lda, B, ldb, C, ldc, M, N, K, bias, act);
        }
    };

    // 1) h = x @ W_in_wrap + b ; LayerNorm
    gemm(x, IN_DIM, W_in_wrap, D_MODEL, h_pre, D_MODEL, NROW, D_MODEL, IN_DIM, b_in_wrap, 0);
    hipLaunchKernelGGL(layernorm_kernel, dim3(NROW), dim3(256), 0, stream,
                       h_pre, ln_gamma, ln_beta);
    // 2) xz = h @ W_in
    gemm(h_pre, D_MODEL, W_in, 2 * D_INNER, xz, 2 * D_INNER, NROW, 2 * D_INNER, D_MODEL, nullptr, 0);
    // 3) u = silu(causal depthwise conv(x_branch))
    hipLaunchKernelGGL(conv_silu_kernel, dim3(NROW * D_INNER / 256), dim3(256), 0, stream,
                       xz, conv_w, conv_b, u);
    // 4) proj = u @ W_x   (cols: [B | C | xdt]); N=2080=65*32 -> TN=2 path
    gemm(u, D_INNER, W_x, W_X_N, proj, W_X_N, NROW, W_X_N, D_INNER, nullptr, 0);
    // 5) dt = softplus(xdt @ W_dt + b_dt)   (xdt starts at col 32 of proj)
    gemm(proj + 2 * D_STATE, W_X_N, W_dt, D_INNER, dtb, D_INNER, NROW, D_INNER, D_INNER, b_dt, 1);
    // 6) selective scan -> y, hT
    hipLaunchKernelGGL(scan_kernel, dim3(BATCH * D_INNER * D_STATE / 256), dim3(256), 0, stream,
                       dtb, u, proj, A_log, h0, yv, hT);
    // 7) yw = (y + u*D_skip) * silu(z)
    hipLaunchKernelGGL(gate_kernel, dim3(NROW * D_INNER / 256), dim3(256), 0, stream,
                       yv, u, xz, D_skip, yw);
    // 8) out = yw @ W_out @ W_out_wrap + b
    gemm(yw, D_INNER, W_out, D_MODEL, t1, D_MODEL, NROW, D_MODEL, D_INNER, nullptr, 0);
    gemm(t1, D_MODEL, W_out_wrap, OUT_DIM, out, OUT_DIM, NROW, OUT_DIM, D_MODEL, b_out_wrap, 0);
}